// DistortionLoss_7928509628908
// MI455X (gfx1250) — compile-verified
//
#include <hip/hip_runtime.h>

// CDNA5 / gfx1250, wave32. Distortion loss (eff_distloss), O(N) cumsum form.
// - Memory-bound: ~403MB streamed -> ~17us floor @ 23.3 TB/s.
// - Software-pipelined async global->LDS staging (ASYNCcnt, in-order
//   completion => counting wait drains the older stage while the newer one
//   is still in flight), ping-pong LDS buffers.
// - Per-ray inclusive scans of w and w*m on the matrix pipe:
//   Incl = W x U (U = upper-triangular ones) via chained
//   V_WMMA_F32_16X16X4_F32 (exact f32), carries through the C operand,
//   chunk totals broadcast with a width-16 shuffle (lowers to DPP row_share).

typedef float v2f __attribute__((ext_vector_type(2)));
typedef float v8f __attribute__((ext_vector_type(8)));

#define N_SAMPLES     128
#define RAYS_PER_TILE 32               // 2 waves x 16 rays
#define THREADS       64               // 2 wave32
#define STAGE_COLS    64               // columns staged per pipeline step
#define BUF_STRIDE    68               // padded floats/row; 68%64=4 -> conflict-free
#define BUF_FLOATS    (RAYS_PER_TILE * BUF_STRIDE)   // per array per buffer
#define ASYNC_PER_STAGE 24             // async b128 ops issued per thread/stage

__device__ __forceinline__ void async_ld16(const float* g, const float* l) {
  // global_load_async_to_lds_b128: VDST = per-lane LDS byte offset (low 32b of
  // the generic pointer), VADDR = per-lane 64-bit global address. ASYNCcnt.
  unsigned lds_off = (unsigned)(unsigned long long)(const void*)l;
  unsigned long long gaddr = (unsigned long long)(const void*)g;
  asm volatile("global_load_async_to_lds_b128 %0, %1, off"
               :: "v"(lds_off), "v"(gaddr) : "memory");
}

template <int CNT>
__device__ __forceinline__ void wait_async() {
#if __has_builtin(__builtin_amdgcn_s_wait_asynccnt)
  __builtin_amdgcn_s_wait_asynccnt(CNT);
#else
  asm volatile("s_wait_asynccnt %0" :: "i"(CNT) : "memory");
#endif
}

__global__ void zero_out_kernel(float* out) { out[0] = 0.0f; }

__global__ void __launch_bounds__(THREADS)
distloss_kernel(const float* __restrict__ W, const float* __restrict__ M,
                const float* __restrict__ S, float* __restrict__ out,
                int numTiles, float scale)
{
  // [2 ping-pong buffers][w|m|s][32 rows x 68 floats] = 52,224 bytes
  __shared__ __align__(16) float sh[2 * 3 * BUF_FLOATS];

  const int t     = threadIdx.x;
  const int lane  = t & 31;
  const int wave  = t >> 5;
  const int half  = lane >> 4;     // 0: lanes 0-15, 1: lanes 16-31
  const int ln    = lane & 15;
  const int kbase = half * 2;      // A/B fragments: lanes16-31 hold K=2,3
  const int wRow  = wave * 16;     // this wave's 16-ray subtile base row

  // Issue one stage (tile, stage) into ping-pong buffer `buf`.
  auto issue = [&](int tile, int stage, int buf) {
    const long long gbase =
        (long long)tile * RAYS_PER_TILE * N_SAMPLES + stage * STAGE_COLS;
    float* dst = sh + buf * 3 * BUF_FLOATS;
    for (int i = t; i < RAYS_PER_TILE * (STAGE_COLS / 4); i += THREADS) {
      const int row = i >> 4;                 // 16 granules per row
      const int c4  = (i & 15) << 2;          // column in floats
      const long long go = gbase + (long long)row * N_SAMPLES + c4;
      const int lo = row * BUF_STRIDE + c4;
      async_ld16(W + go, dst + lo);
      async_ld16(M + go, dst + BUF_FLOATS + lo);
      async_ld16(S + go, dst + 2 * BUF_FLOATS + lo);
    }
  };

  float acc = 0.0f;
  int buf = 0;

  // Prologue: prime the pipeline with (first tile, stage 0).
  issue(blockIdx.x, 0, 0);

  for (int tile = blockIdx.x; tile < numTiles; tile += gridDim.x) {
    v8f carryW  = {0,0,0,0,0,0,0,0};     // C/D layout: comp r, lane n
    v8f carryWM = {0,0,0,0,0,0,0,0};

    #pragma unroll 1
    for (int stage = 0; stage < 2; ++stage) {
      // Prefetch the next stage into the other buffer, then drain the
      // current stage's 24 older async ops (in-order completion).
      const int ntile  = (stage == 0) ? tile : tile + (int)gridDim.x;
      const int nstage = stage ^ 1;
      if (ntile < numTiles) {
        issue(ntile, nstage, buf ^ 1);
        wait_async<ASYNC_PER_STAGE>();
      } else {
        wait_async<0>();
      }
      __syncthreads();

      const float* bw = sh + buf * 3 * BUF_FLOATS;
      const float* bm = bw + BUF_FLOATS;
      const float* bs = bm + BUF_FLOATS;
      const int arow = wRow + ln;            // A layout: lane -> row M

      #pragma unroll 1
      for (int c = 0; c < STAGE_COLS / 16; ++c) {   // 4 chunks of 16 cols
        const int colBase = c * 16;
        v8f cw  = carryW;                    // carry enters through C operand
        v8f cwm = carryWM;
        #pragma unroll
        for (int q = 0; q < 4; ++q) {        // K = 4 per WMMA, chained -> K=16
          const int colOff = colBase + q * 4 + kbase;
          v2f aw  = *(const v2f*)&bw[arow * BUF_STRIDE + colOff];
          v2f am  = *(const v2f*)&bm[arow * BUF_STRIDE + colOff];
          v2f awm = aw * am;                 // w*m fragment
          v2f bu;                            // U[k,j] = (k <= j)
          bu.x = (q * 4 + kbase     <= ln) ? 1.0f : 0.0f;
          bu.y = (q * 4 + kbase + 1 <= ln) ? 1.0f : 0.0f;
          cw  = __builtin_amdgcn_wmma_f32_16x16x4_f32(false, aw,  false, bu,
                                                      (short)0, cw,  false, false);
          cwm = __builtin_amdgcn_wmma_f32_16x16x4_f32(false, awm, false, bu,
                                                      (short)0, cwm, false, false);
        }

        // Elementwise terms in D layout; update carries (DPP row_share).
        const int col = colBase + ln;
        #pragma unroll
        for (int r = 0; r < 8; ++r) {
          const int row = wRow + half * 8 + r;   // D: comp r -> M = r (+8)
          const float wv  = bw[row * BUF_STRIDE + col];
          const float mv  = bm[row * BUF_STRIDE + col];
          const float sv  = bs[row * BUF_STRIDE + col];
          const float wmv = wv * mv;
          const float wex  = cw[r]  - wv;        // exclusive prefix of w
          const float wmex = cwm[r] - wmv;       // exclusive prefix of w*m
          acc += (1.0f / 3.0f) * sv * wv * wv
               + 2.0f * (wmv * wex - wv * wmex);
          carryW[r]  = __shfl(cw[r],  15, 16);   // chunk totals -> next C
          carryWM[r] = __shfl(cwm[r], 15, 16);
        }
      }

      __syncthreads();   // everyone done reading buf before it is re-issued
      buf ^= 1;
    }
  }

  // ---- reduce: wave32 shuffle tree, one f32 atomic per wave ----
  float v = acc * scale;
  #pragma unroll
  for (int off = 16; off > 0; off >>= 1) v += __shfl_down(v, off, 32);
  if (lane == 0) atomicAdd(out, v);
}

extern "C" void kernel_launch(void* const* d_in, const int* in_sizes, int n_in,
                              void* d_out, int out_size, void* d_ws, size_t ws_size,
                              hipStream_t stream) {
  const float* W = (const float*)d_in[0];   // weights  [B,128]
  const float* M = (const float*)d_in[1];   // distances[B,128]
  const float* S = (const float*)d_in[2];   // intervals[B,128]
  float* out = (float*)d_out;

  const long long total = (long long)in_sizes[0];
  const int B = (int)(total / N_SAMPLES);
  const int numTiles = B / RAYS_PER_TILE;   // 8192 for B=262144
  const float scale = 0.01f / (float)B;     // LOSS_WEIGHT * mean

  int blocks = numTiles < 2048 ? numTiles : 2048;

  hipLaunchKernelGGL(zero_out_kernel, dim3(1), dim3(1), 0, stream, out);
  hipLaunchKernelGGL(distloss_kernel, dim3(blocks), dim3(THREADS), 0, stream,
                     W, M, S, out, numTiles, scale);
}